// mLSTMBlock_20031727468598
// MI455X (gfx1250) — compile-verified
//
#include <hip/hip_runtime.h>
#include <hip/hip_bf16.h>
#include <math.h>

// mLSTM block for MI455X (gfx1250), fp32 end-to-end via V_WMMA_F32_16X16X4_F32.
// B = H = 96, D = 768, H2 = 9216, kqv row stride = 3*H2 = 27648.
// Register-blocked: 4 N-tiles per wave share each A load; denom kernel hoists
// the whole K=96 A-tile into 48 VGPRs and sweeps 16 L-tiles against it.

typedef __attribute__((ext_vector_type(2))) float v2f;
typedef __attribute__((ext_vector_type(8))) float v8f;

#define BDIM   96
#define HDIM   96
#define DDIM   768
#define H2DIM  9216
#define KQVLD  27648   // 3*H2

static __device__ __forceinline__ v8f wmma4(v2f a, v2f b, v8f c) {
  // D = A(16x4,f32) * B(4x16,f32) + C(16x16,f32)
  return __builtin_amdgcn_wmma_f32_16x16x4_f32(
      /*neg_a=*/false, a, /*neg_b=*/false, b,
      /*c_mod=*/(short)0, c, /*reuse_a=*/false, /*reuse_b=*/false);
}

// ---- NT adjacent 16x16 output tiles sharing one A stream ------------------
// A element (m,k) at A[m*a_rs + k*a_cs]; B element (k,n) at Bm[k*b_rs + n*b_cs].
// Tile t covers columns [n0 + 16*t, n0 + 16*t + 16).
template <int NT>
static __device__ __forceinline__ void mm_tileN(
    const float* __restrict__ A, long a_rs, long a_cs,
    const float* __restrict__ Bm, long b_rs, long b_cs,
    int m0, int n0, int kbeg, int kend, v8f* acc) {
  const int lane = (int)(threadIdx.x & 31u);
  const int half = lane >> 4;       // which K-pair this lane feeds
  const int l16  = lane & 15;       // row of A / col of B this lane owns
  const float* ap = A + (long)(m0 + l16) * a_rs;
  const float* bp[NT];
#pragma unroll
  for (int t = 0; t < NT; ++t) bp[t] = Bm + (long)(n0 + 16 * t + l16) * b_cs;
#pragma unroll 4
  for (int k = kbeg; k < kend; k += 4) {
    const int kk = k + 2 * half;
    v2f a;
    a[0] = ap[(long)kk * a_cs];
    a[1] = ap[(long)(kk + 1) * a_cs];
#pragma unroll
    for (int t = 0; t < NT; ++t) {
      v2f b;
      b[0] = bp[t][(long)kk * b_rs];
      b[1] = bp[t][(long)(kk + 1) * b_rs];
      acc[t] = wmma4(a, b, acc[t]);
    }
  }
}

static __device__ __forceinline__ void store_tile(float* __restrict__ out, long ldo,
                                                  int m0, int n0, v8f c) {
  const int lane = (int)(threadIdx.x & 31u);
  const int half = lane >> 4, l16 = lane & 15;
#pragma unroll
  for (int r = 0; r < 8; ++r)
    out[(long)(m0 + r + 8 * half) * ldo + n0 + l16] = c[r];
}

// ---- Stage 1: i = exp(x wix^T + b), f = exp(h_prev wfx^T + b) -------------
__global__ __launch_bounds__(32) void k_gates(
    const float* __restrict__ x, const float* __restrict__ h_prev,
    const float* __restrict__ wix_w, const float* __restrict__ wix_b,
    const float* __restrict__ wfx_w, const float* __restrict__ wfx_b,
    float* __restrict__ ws_i, float* __restrict__ ws_f) {
  const int m0 = blockIdx.x * 16, n0 = blockIdx.y * 16;
  const int lane = (int)(threadIdx.x & 31u);
  const int half = lane >> 4, l16 = lane & 15;
  v8f acc[1] = {};
  if (blockIdx.z == 0) {
    mm_tileN<1>(x, DDIM, 1, wix_w, /*b_rs=*/1, /*b_cs=*/DDIM, m0, n0, 0, DDIM, acc);
    const float bias = wix_b[n0 + l16];
#pragma unroll
    for (int r = 0; r < 8; ++r)
      ws_i[(long)(m0 + r + 8 * half) * HDIM + n0 + l16] = expf(acc[0][r] + bias);
  } else {
    mm_tileN<1>(h_prev, HDIM, 1, wfx_w, 1, HDIM, m0, n0, 0, HDIM, acc);
    const float bias = wfx_b[n0 + l16];
#pragma unroll
    for (int r = 0; r < 8; ++r)
      ws_f[(long)(m0 + r + 8 * half) * HDIM + n0 + l16] = expf(acc[0][r] + bias);
  }
}

// ---- Stage 2: kqv = x Wkqv^T + b ; k-slice pre-scaled by 1/sqrt(H2) -------
// 4 waves/block, each wave owns a 16x64 strip (4 N-tiles share the A stream).
__global__ __launch_bounds__(128) void k_kqv(
    const float* __restrict__ x, const float* __restrict__ W,
    const float* __restrict__ bias, float* __restrict__ kqv) {
  const int w  = (int)(threadIdx.x >> 5u);
  const int n0 = blockIdx.x * 256 + w * 64;
  const int m0 = blockIdx.y * 16;
  const int lane = (int)(threadIdx.x & 31u);
  const int half = lane >> 4, l16 = lane & 15;
  __builtin_prefetch(W + (long)(n0 + l16) * DDIM, 0, 1);  // stream the weight rows
  v8f acc[4] = {};
  mm_tileN<4>(x, DDIM, 1, W, 1, DDIM, m0, n0, 0, DDIM, acc);
#pragma unroll
  for (int t = 0; t < 4; ++t) {
    const int n = n0 + 16 * t + l16;
    const float b = bias[n];
    const float scale = (n < H2DIM) ? (1.0f / 96.0f) : 1.0f;  // sqrt(9216)=96
#pragma unroll
    for (int r = 0; r < 8; ++r)
      kqv[(long)(m0 + r + 8 * half) * KQVLD + n] = (acc[t][r] + b) * scale;
  }
}

// ---- Stage 3: o = sigmoid(h_prev Wox^T + b) -------------------------------
__global__ __launch_bounds__(128) void k_ogate(
    const float* __restrict__ h_prev, const float* __restrict__ W,
    const float* __restrict__ bias, float* __restrict__ o) {
  const int w  = (int)(threadIdx.x >> 5u);
  const int n0 = blockIdx.x * 256 + w * 64;
  const int m0 = blockIdx.y * 16;
  const int lane = (int)(threadIdx.x & 31u);
  const int half = lane >> 4, l16 = lane & 15;
  v8f acc[4] = {};
  mm_tileN<4>(h_prev, HDIM, 1, W, 1, HDIM, m0, n0, 0, HDIM, acc);
#pragma unroll
  for (int t = 0; t < 4; ++t) {
    const float b = bias[n0 + 16 * t + l16];
#pragma unroll
    for (int r = 0; r < 8; ++r) {
      const float z = acc[t][r] + b;
      o[(long)(m0 + r + 8 * half) * H2DIM + n0 + 16 * t + l16] =
          1.0f / (1.0f + expf(-z));
    }
  }
}

// ---- Stage 4: vkT = v k^T  (split-K across 8 waves, LDS reduce) -----------
__global__ __launch_bounds__(256) void k_vkt(const float* __restrict__ kqv,
                                             float* __restrict__ vkT) {
  __shared__ float red[8][32][8];
  const int w    = (int)(threadIdx.x >> 5u);
  const int lane = (int)(threadIdx.x & 31u);
  const int m0 = blockIdx.x * 16, n0 = blockIdx.y * 16;
  const int kbeg = w * (H2DIM / 8), kend = kbeg + (H2DIM / 8);
  // A = v (rows of kqv at col offset 2*H2); B(j,b') = k[b'][j] -> b_cs=KQVLD
  v8f acc[1] = {};
  mm_tileN<1>(kqv + 2 * H2DIM, KQVLD, 1, kqv, /*b_rs=*/1, /*b_cs=*/KQVLD,
              m0, n0, kbeg, kend, acc);
#pragma unroll
  for (int r = 0; r < 8; ++r) red[w][lane][r] = acc[0][r];
  __syncthreads();
  if (w == 0) {
    const int half = lane >> 4, l16 = lane & 15;
#pragma unroll
    for (int r = 0; r < 8; ++r) {
      float s = 0.0f;
#pragma unroll
      for (int ww = 0; ww < 8; ++ww) s += red[ww][lane][r];
      vkT[(long)(m0 + r + 8 * half) * HDIM + n0 + l16] = s;
    }
  }
}

// ---- Stage 5: C = f C_prev + i vkT ---------------------------------------
__global__ __launch_bounds__(32) void k_cell(
    const float* __restrict__ ws_f, const float* __restrict__ C_prev,
    const float* __restrict__ ws_i, const float* __restrict__ vkT,
    float* __restrict__ C_out) {
  const int m0 = blockIdx.x * 16, n0 = blockIdx.y * 16;
  v8f acc[1] = {};
  mm_tileN<1>(ws_f, HDIM, 1, C_prev, HDIM, 1, m0, n0, 0, HDIM, acc);
  mm_tileN<1>(ws_i, HDIM, 1, vkT,   HDIM, 1, m0, n0, 0, HDIM, acc);
  store_tile(C_out, HDIM, m0, n0, acc[0]);
}

// ---- Stage 6: n = f n_prev + i k -----------------------------------------
__global__ __launch_bounds__(128) void k_norm(
    const float* __restrict__ ws_f, const float* __restrict__ n_prev,
    const float* __restrict__ ws_i, const float* __restrict__ kqv,
    float* __restrict__ n_out) {
  const int w  = (int)(threadIdx.x >> 5u);
  const int n0 = blockIdx.x * 256 + w * 64;
  const int m0 = blockIdx.y * 16;
  v8f acc[4] = {};
  mm_tileN<4>(ws_f, HDIM, 1, n_prev, H2DIM, 1, m0, n0, 0, HDIM, acc);
  mm_tileN<4>(ws_i, HDIM, 1, kqv,    KQVLD, 1, m0, n0, 0, HDIM, acc);  // k slice
#pragma unroll
  for (int t = 0; t < 4; ++t) store_tile(n_out, H2DIM, m0, n0 + 16 * t, acc[t]);
}

// ---- Stage 7a: zero denom (abs values >= 0, so uint bit-order == order) ---
__global__ void k_zero(unsigned int* __restrict__ p, int n) {
  const int i = blockIdx.x * blockDim.x + threadIdx.x;
  if (i < n) p[i] = 0u;
}

// ---- Stage 7b: denom[j] = max_l |(n^T q)[j,l]| ----------------------------
// A-tile of n^T (16 rows j, K=96) hoisted into 48 VGPRs once per wave, then
// 16 L-tiles (4 groups x 4-way N blocking) swept against it: 384 WMMA/wave
// with zero A reloads.
__global__ __launch_bounds__(32) void k_denom(
    const float* __restrict__ nmat, const float* __restrict__ kqv,
    unsigned int* __restrict__ denom) {
  const float* q = kqv + H2DIM;
  const int j0    = blockIdx.x * 16;
  const int lbase = blockIdx.y * 256;   // 256 l-values per wave
  const int lane = (int)(threadIdx.x & 31u);
  const int half = lane >> 4, l16 = lane & 15;

  // A(j,b) = n[b][j]: lanes span 16 consecutive j -> coalesced row reads.
  float areg[48];
  {
    const float* ap = nmat + j0 + l16;
#pragma unroll
    for (int t = 0; t < 24; ++t) {
      const int kk = 4 * t + 2 * half;
      areg[2 * t]     = ap[(long)kk * H2DIM];
      areg[2 * t + 1] = ap[(long)(kk + 1) * H2DIM];
    }
  }

  float rowmax[8];
#pragma unroll
  for (int r = 0; r < 8; ++r) rowmax[r] = 0.0f;

  for (int g = 0; g < 4; ++g) {
    const float* bp = q + lbase + g * 64 + l16;  // B(b,l) = q[b][l]
    v8f acc[4] = {};
#pragma unroll
    for (int t = 0; t < 24; ++t) {
      const int kk = 4 * t + 2 * half;
      v2f a;
      a[0] = areg[2 * t];
      a[1] = areg[2 * t + 1];
#pragma unroll
      for (int u = 0; u < 4; ++u) {
        v2f b;
        b[0] = bp[(long)kk * KQVLD + 16 * u];
        b[1] = bp[(long)(kk + 1) * KQVLD + 16 * u];
        acc[u] = wmma4(a, b, acc[u]);
      }
    }
#pragma unroll
    for (int u = 0; u < 4; ++u)
#pragma unroll
      for (int r = 0; r < 8; ++r)
        rowmax[r] = fmaxf(rowmax[r], fabsf(acc[u][r]));
  }

  // max over tile columns: lanes 0-15 and 16-31 reduce separately (xor<16)
#pragma unroll
  for (int s = 1; s < 16; s <<= 1)
#pragma unroll
    for (int r = 0; r < 8; ++r)
      rowmax[r] = fmaxf(rowmax[r], __shfl_xor(rowmax[r], s, 32));
  if (l16 == 0) {
#pragma unroll
    for (int r = 0; r < 8; ++r)
      atomicMax(&denom[j0 + r + 8 * half], __float_as_uint(rowmax[r]));
  }
}

// ---- Stage 8: h = o * (C q) / denom ---------------------------------------
__global__ __launch_bounds__(128) void k_hout(
    const float* __restrict__ Cmat, const float* __restrict__ kqv,
    const float* __restrict__ o, const unsigned int* __restrict__ denom,
    float* __restrict__ h) {
  const float* q = kqv + H2DIM;
  const int w  = (int)(threadIdx.x >> 5u);
  const int n0 = blockIdx.x * 256 + w * 64;
  const int m0 = blockIdx.y * 16;
  const int lane = (int)(threadIdx.x & 31u);
  const int half = lane >> 4, l16 = lane & 15;
  v8f acc[4] = {};
  mm_tileN<4>(Cmat, HDIM, 1, q, /*b_rs=*/KQVLD, /*b_cs=*/1, m0, n0, 0, HDIM, acc);
#pragma unroll
  for (int t = 0; t < 4; ++t) {
    const int n = n0 + 16 * t + l16;
    const float dinv = 1.0f / __uint_as_float(denom[n]);
#pragma unroll
    for (int r = 0; r < 8; ++r) {
      const long row = m0 + r + 8 * half;
      h[row * H2DIM + n] = o[row * H2DIM + n] * acc[t][r] * dinv;
    }
  }
}

extern "C" void kernel_launch(void* const* d_in, const int* in_sizes, int n_in,
                              void* d_out, int out_size, void* d_ws, size_t ws_size,
                              hipStream_t stream) {
  const float* x      = (const float*)d_in[0];
  const float* h_prev = (const float*)d_in[1];
  const float* C_prev = (const float*)d_in[2];
  const float* n_prev = (const float*)d_in[3];
  const float* wix_w  = (const float*)d_in[4];
  const float* wix_b  = (const float*)d_in[5];
  const float* wfx_w  = (const float*)d_in[6];
  const float* wfx_b  = (const float*)d_in[7];
  const float* Wox_w  = (const float*)d_in[8];
  const float* Wox_b  = (const float*)d_in[9];
  const float* Wkqv_w = (const float*)d_in[10];
  const float* Wkqv_b = (const float*)d_in[11];

  float* out   = (float*)d_out;
  float* h_out = out;                         // [96, 9216]
  float* C_out = out + 96 * 9216;             // [96, 96]
  float* n_out = out + 96 * 9216 + 96 * 96;   // [96, 9216]

  float* ws = (float*)d_ws;
  float*        ws_i   = ws;                  //   9216
  float*        ws_f   = ws + 9216;           //   9216
  float*        ws_vkt = ws + 18432;          //   9216
  unsigned int* ws_den = (unsigned int*)(ws + 27648);  // 9216
  float*        ws_o   = ws + 36864;          // 884736
  float*        ws_kqv = ws + 921600;         // 2654208  (total ~14.3 MB)

  k_gates<<<dim3(6, 6, 2),  32, 0, stream>>>(x, h_prev, wix_w, wix_b, wfx_w, wfx_b, ws_i, ws_f);
  k_kqv  <<<dim3(108, 6),  128, 0, stream>>>(x, Wkqv_w, Wkqv_b, ws_kqv);
  k_ogate<<<dim3(36, 6),   128, 0, stream>>>(h_prev, Wox_w, Wox_b, ws_o);
  k_vkt  <<<dim3(6, 6),    256, 0, stream>>>(ws_kqv, ws_vkt);
  k_cell <<<dim3(6, 6),     32, 0, stream>>>(ws_f, C_prev, ws_i, ws_vkt, C_out);
  k_norm <<<dim3(36, 6),   128, 0, stream>>>(ws_f, n_prev, ws_i, ws_kqv, n_out);
  k_zero <<<36,            256, 0, stream>>>(ws_den, 9216);
  k_denom<<<dim3(576, 36),  32, 0, stream>>>(n_out, ws_kqv, ws_den);
  k_hout <<<dim3(36, 6),   128, 0, stream>>>(C_out, ws_kqv, ws_o, ws_den, h_out);
}